// Aggregator_9414568312928
// MI455X (gfx1250) — compile-verified
//
#include <hip/hip_runtime.h>
#include <hip/hip_bf16.h>
#include <math.h>

#define MIN_NORM 1e-15f
#define D 64

typedef __attribute__((ext_vector_type(2))) float v2f;
typedef __attribute__((ext_vector_type(8))) float v8f;

__device__ __forceinline__ float wred(float v) {
#pragma unroll
    for (int off = 16; off > 0; off >>= 1)
        v += __shfl_xor(v, off, 32);
    return v;
}

// ---------------------------------------------------------------------------
// Zero init: d_out accumulation regions + cnt[] in workspace
// ---------------------------------------------------------------------------
__global__ void zero_kernel(float* __restrict__ out, long n_out,
                            float* __restrict__ cnt, long n_cnt) {
    long stride = (long)gridDim.x * blockDim.x;
    long total = n_out + n_cnt;
    for (long i = (long)blockIdx.x * blockDim.x + threadIdx.x; i < total; i += stride) {
        if (i < n_out) out[i] = 0.0f;
        else           cnt[i - n_out] = 0.0f;
    }
}

// ---------------------------------------------------------------------------
// Edge kernel: one wave32 per edge, 2 floats per lane.
// Computes logmap(project(mobius_add(hyper_tail, hyper_rel)), hyper_head)
// and atomically accumulates into entity_acc[head], cnt[head].
// ---------------------------------------------------------------------------
__global__ void edge_kernel(const float* __restrict__ ent,
                            const float* __restrict__ relw,
                            const int*   __restrict__ eidx,
                            const int*   __restrict__ etype,
                            float* __restrict__ eacc,
                            float* __restrict__ cnt,
                            int E) {
    int wave = (int)((blockIdx.x * (long)blockDim.x + threadIdx.x) >> 5);
    if (wave >= E) return;                  // wave-uniform exit
    int lane = threadIdx.x & 31;

    int head = eidx[wave];
    int tail = eidx[E + wave];
    int rel  = etype[wave] - 1;

    float2 h = ((const float2*)(ent  + (size_t)head * D))[lane];
    float2 t = ((const float2*)(ent  + (size_t)tail * D))[lane];
    float2 r = ((const float2*)(relw + (size_t)rel  * D))[lane];

    // hyper_head = expmap0(h)
    float h2 = wred(h.x * h.x + h.y * h.y);
    float nh = fmaxf(sqrtf(h2), MIN_NORM);
    float th = tanhf(nh);
    float sp = th / nh;
    float2 p = {sp * h.x, sp * h.y};
    float p2 = (th * th) * (h2 / (nh * nh));          // ||p||^2 (exact)
    float lam = 2.0f / fmaxf(1.0f - p2, MIN_NORM);

    // gamma_t = tanh(0.5*lam*||t||) * t/||t||;  hyper_tail = mobius_add(p, gamma_t)
    float t2 = wred(t.x * t.x + t.y * t.y);
    float nt = fmaxf(sqrtf(t2), MIN_NORM);
    float st = tanhf(0.5f * lam * nt);
    float sg = st / nt;
    float2 gt = {sg * t.x, sg * t.y};
    float gt2 = st * st * (t2 / (nt * nt));
    float pgt = wred(p.x * gt.x + p.y * gt.y);
    float a1 = 1.0f + 2.0f * pgt + gt2;
    float a2 = 1.0f - p2;
    float iad = 1.0f / fmaxf(1.0f + 2.0f * pgt + p2 * gt2, MIN_NORM);
    float2 ht = {(a1 * p.x + a2 * gt.x) * iad, (a1 * p.y + a2 * gt.y) * iad};
    float ht2 = (a1 * a1 * p2 + 2.0f * a1 * a2 * pgt + a2 * a2 * gt2) * iad * iad;

    // gamma_r ; hyper_rel = mobius_add(p, gamma_r)
    float r2 = wred(r.x * r.x + r.y * r.y);
    float nr = fmaxf(sqrtf(r2), MIN_NORM);
    float sr = tanhf(0.5f * lam * nr);
    float sg2 = sr / nr;
    float2 gr = {sg2 * r.x, sg2 * r.y};
    float gr2 = sr * sr * (r2 / (nr * nr));
    float pgr = wred(p.x * gr.x + p.y * gr.y);
    float b1 = 1.0f + 2.0f * pgr + gr2;
    float b2 = 1.0f - p2;
    float ibd = 1.0f / fmaxf(1.0f + 2.0f * pgr + p2 * gr2, MIN_NORM);
    float2 hr = {(b1 * p.x + b2 * gr.x) * ibd, (b1 * p.y + b2 * gr.y) * ibd};
    float hr2 = (b1 * b1 * p2 + 2.0f * b1 * b2 * pgr + b2 * b2 * gr2) * ibd * ibd;

    // m = mobius_add(ht, hr); ||m||^2 in closed form
    float hthr = wred(ht.x * hr.x + ht.y * hr.y);
    float c1 = 1.0f + 2.0f * hthr + hr2;
    float c2 = 1.0f - ht2;
    float icd = 1.0f / fmaxf(1.0f + 2.0f * hthr + ht2 * hr2, MIN_NORM);
    float2 m = {(c1 * ht.x + c2 * hr.x) * icd, (c1 * ht.y + c2 * hr.y) * icd};
    float m2 = (c1 * c1 * ht2 + 2.0f * c1 * c2 * hthr + c2 * c2 * hr2) * icd * icd;

    // project to ball
    float nm = fmaxf(sqrtf(m2), MIN_NORM);
    const float maxn = 1.0f - 1e-5f;
    if (nm > maxn) {
        float s = maxn / nm;
        m.x *= s; m.y *= s;
        m2 = maxn * maxn;
    }

    // logmap(m, p): sub = mobius_add(-p, m)
    float pm = wred(p.x * m.x + p.y * m.y);
    float xy = -pm;                                    // dot(-p, m)
    float e1 = 1.0f + 2.0f * xy + m2;                  // coeff of (-p)
    float e2 = 1.0f - p2;                              // coeff of m
    float ied = 1.0f / fmaxf(1.0f + 2.0f * xy + p2 * m2, MIN_NORM);
    float2 sub = {(-e1 * p.x + e2 * m.x) * ied, (-e1 * p.y + e2 * m.y) * ied};
    float sub2 = (e1 * e1 * p2 + 2.0f * e1 * e2 * xy + e2 * e2 * m2) * ied * ied;
    float ns = fmaxf(sqrtf(sub2), MIN_NORM);
    float arg = fminf(fmaxf(ns, MIN_NORM), 1.0f - 1e-5f);
    float scale = (2.0f / lam) * atanhf(arg) / ns;

    float* dst = eacc + (size_t)head * D + lane * 2;
    atomicAdd(dst,     scale * sub.x);
    atomicAdd(dst + 1, scale * sub.y);
    if (lane == 0) atomicAdd(cnt + head, 1.0f);
}

// ---------------------------------------------------------------------------
// entity_agg = ssum / max(cnt, 1)
// ---------------------------------------------------------------------------
__global__ void div_kernel(float* __restrict__ eagg,
                           const float* __restrict__ cnt, long total) {
    long i = (long)blockIdx.x * blockDim.x + threadIdx.x;
    if (i >= total) return;
    eagg[i] /= fmaxf(cnt[i >> 6], 1.0f);
}

// ---------------------------------------------------------------------------
// item_agg_cf[mat_col] += user_emb[mat_row]  (one wave per nnz)
// ---------------------------------------------------------------------------
__global__ void item_agg_kernel(const float* __restrict__ user_emb,
                                const int* __restrict__ mrow,
                                const int* __restrict__ mcol,
                                float* __restrict__ item_agg, int nnz) {
    int w = (int)((blockIdx.x * (long)blockDim.x + threadIdx.x) >> 5);
    if (w >= nnz) return;
    int lane = threadIdx.x & 31;
    int u  = mrow[w];
    int it = mcol[w];
    float2 v = ((const float2*)(user_emb + (size_t)u * D))[lane];
    float* dst = item_agg + (size_t)it * D + lane * 2;
    atomicAdd(dst,     v.x);
    atomicAdd(dst + 1, v.y);
}

// ---------------------------------------------------------------------------
// Fusion GEMM: gi = sigmoid(cf@W1^T + kg@W2^T); fusion = gi*cf + (1-gi)*kg
// fp32 WMMA 16x16x4: one wave per 16x16 output tile, K=64 unrolled.
// A[lane] layout: lanes 0-15 hold M=lane; VGPR0/1 = K={k0,k0+1} (lanes<16)
//                 or K={k0+2,k0+3} (lanes>=16)  -> float2 at A[row, k0+2*(lane>>4)]
// B (4x16, K-major): vgpr pair = W[n, k0+2*(lane>>4) + {0,1}], n = lane&15 (+16 split)
// ---------------------------------------------------------------------------
__global__ void fusion_gemm_kernel(const float* __restrict__ cf,
                                   const float* __restrict__ kg,
                                   const float* __restrict__ W1,
                                   const float* __restrict__ W2,
                                   float* __restrict__ fusion,
                                   int n_items) {
    int wave = (int)((blockIdx.x * (long)blockDim.x + threadIdx.x) >> 5);
    int mtiles = n_items >> 4;
    int mt = wave >> 2;           // 4 N-tiles of 16 cover D=64
    int nt = wave & 3;
    if (mt >= mtiles) return;     // wave-uniform exit, EXEC all-ones inside

    int lane = threadIdx.x & 31;
    int half = lane >> 4;         // 0 or 1
    int l16  = lane & 15;

    const float* aCf = cf + (size_t)(mt * 16 + l16) * D + half * 2;
    const float* aKg = kg + (size_t)(mt * 16 + l16) * D + half * 2;
    const float* b1p = W1 + (size_t)(nt * 16 + l16) * D + half * 2;
    const float* b2p = W2 + (size_t)(nt * 16 + l16) * D + half * 2;

    v8f acc = {};
#pragma unroll
    for (int k = 0; k < D; k += 4) {
        v2f a1 = *(const v2f*)(aCf + k);
        v2f w1 = *(const v2f*)(b1p + k);
        v2f a2 = *(const v2f*)(aKg + k);
        v2f w2 = *(const v2f*)(b2p + k);
        acc = __builtin_amdgcn_wmma_f32_16x16x4_f32(false, a1, false, w1,
                                                    (short)0, acc, false, false);
        acc = __builtin_amdgcn_wmma_f32_16x16x4_f32(false, a2, false, w2,
                                                    (short)0, acc, false, false);
    }

    // C/D layout: vgpr g -> M = g (lanes 0-15) or g+8 (lanes 16-31), N = lane&15
    int ccol  = nt * 16 + l16;
    int crow0 = mt * 16 + half * 8;
#pragma unroll
    for (int g = 0; g < 8; ++g) {
        int row = crow0 + g;
        float cfe = cf[(size_t)row * D + ccol];
        float kge = kg[(size_t)row * D + ccol];
        float gi = 1.0f / (1.0f + expf(-acc[g]));
        fusion[(size_t)row * D + ccol] = gi * cfe + (1.0f - gi) * kge;
    }
}

// ---------------------------------------------------------------------------
// user_agg[mat_row] += mat_val * fusion[mat_col]  (one wave per nnz)
// ---------------------------------------------------------------------------
__global__ void user_agg_kernel(const float* __restrict__ fusion,
                                const float* __restrict__ mval,
                                const int* __restrict__ mrow,
                                const int* __restrict__ mcol,
                                float* __restrict__ user_agg, int nnz) {
    int w = (int)((blockIdx.x * (long)blockDim.x + threadIdx.x) >> 5);
    if (w >= nnz) return;
    int lane = threadIdx.x & 31;
    int u  = mrow[w];
    int it = mcol[w];
    float val = mval[w];
    float2 f = ((const float2*)(fusion + (size_t)it * D))[lane];
    float* dst = user_agg + (size_t)u * D + lane * 2;
    atomicAdd(dst,     val * f.x);
    atomicAdd(dst + 1, val * f.y);
}

// ---------------------------------------------------------------------------
extern "C" void kernel_launch(void* const* d_in, const int* in_sizes, int n_in,
                              void* d_out, int out_size, void* d_ws, size_t ws_size,
                              hipStream_t stream) {
    const float* entity_emb      = (const float*)d_in[0];
    const float* user_emb        = (const float*)d_in[1];
    const float* item_emb_cf     = (const float*)d_in[2];
    const float* relation_weight = (const float*)d_in[3];
    const float* W1              = (const float*)d_in[4];
    const float* W2              = (const float*)d_in[5];
    const float* mat_val         = (const float*)d_in[6];
    const int*   edge_index      = (const int*)d_in[7];
    const int*   edge_type       = (const int*)d_in[8];
    const int*   mat_row         = (const int*)d_in[9];
    const int*   mat_col         = (const int*)d_in[10];

    const int n_ent   = in_sizes[0] / D;
    const int n_users = in_sizes[1] / D;
    const int n_items = in_sizes[2] / D;
    const int E       = in_sizes[7] / 2;
    const int nnz     = in_sizes[9];

    float* out        = (float*)d_out;
    float* entity_agg = out;                                  // [n_ent  x 64]
    float* user_agg   = out + (size_t)n_ent * D;              // [n_users x 64]
    float* item_agg   = user_agg + (size_t)n_users * D;       // [n_items x 64]

    float* cnt    = (float*)d_ws;                             // [n_ent]
    float* fusion = cnt + n_ent;                              // [n_items x 64]

    // 1) zero accumulators
    {
        long total = (long)out_size + n_ent;
        int blocks = (int)((total + 255) / 256);
        zero_kernel<<<blocks, 256, 0, stream>>>(out, (long)out_size, cnt, (long)n_ent);
    }
    // 2) edge aggregation (one wave per edge)
    {
        int blocks = (E + 7) / 8;  // 8 waves per 256-thread block
        edge_kernel<<<blocks, 256, 0, stream>>>(entity_emb, relation_weight,
                                                edge_index, edge_type,
                                                entity_agg, cnt, E);
    }
    // 3) entity_agg /= max(cnt, 1)
    {
        long total = (long)n_ent * D;
        int blocks = (int)((total + 255) / 256);
        div_kernel<<<blocks, 256, 0, stream>>>(entity_agg, cnt, total);
    }
    // 4) item_agg_cf scatter
    {
        int blocks = (nnz + 7) / 8;
        item_agg_kernel<<<blocks, 256, 0, stream>>>(user_emb, mat_row, mat_col,
                                                    item_agg, nnz);
    }
    // 5) fusion GEMM (fp32 WMMA)
    {
        int tiles = (n_items / 16) * 4;
        int blocks = (tiles + 7) / 8;
        fusion_gemm_kernel<<<blocks, 256, 0, stream>>>(item_emb_cf, entity_emb,
                                                       W1, W2, fusion, n_items);
    }
    // 6) user_agg scatter
    {
        int blocks = (nnz + 7) / 8;
        user_agg_kernel<<<blocks, 256, 0, stream>>>(fusion, mat_val, mat_row,
                                                    mat_col, user_agg, nnz);
    }
}